// ModelTorch_83038897701196
// MI455X (gfx1250) — compile-verified
//
#include <hip/hip_runtime.h>
#include <math.h>

typedef __attribute__((ext_vector_type(2))) float v2f;
typedef __attribute__((ext_vector_type(8))) float v8f;

#define NWAVES 8
#define TILE   16          // keys per WMMA tile
#define DHEAD  128
#define SLEN   4096
#define HKV    8
#define GQA    4
#define CHUNK  (SLEN / NWAVES)   // 512 keys per wave
#define NTILES (CHUNK / TILE)    // 32 tiles per wave

__launch_bounds__(256)
__global__ void gqa_decode_fused(const float* __restrict__ q,
                                 const float* __restrict__ K,
                                 const float* __restrict__ V,
                                 float* __restrict__ out)
{
    // per-wave P transpose scratch + cross-wave combine buffers
    __shared__ __align__(16) float P_lds[NWAVES][GQA][TILE];
    __shared__ __align__(16) float O_lds[NWAVES][GQA][DHEAD];
    __shared__ float m_lds[NWAVES][GQA];
    __shared__ float l_lds[NWAVES][GQA];

    const int tid  = threadIdx.x;
    const int lane = tid & 31;
    const int w    = tid >> 5;
    const int half = lane >> 4;   // which 16-lane half
    const int lrow = lane & 15;

    const int b   = blockIdx.x / HKV;
    const int hkv = blockIdx.x % HKV;

    const float scale = 0.08838834764831845f; // 1/sqrt(128)

    const float* Kb = K + ((size_t)(b * HKV + hkv)) * SLEN * DHEAD;
    const float* Vb = V + ((size_t)(b * HKV + hkv)) * SLEN * DHEAD;
    const float* qb = q + ((size_t)(b * HKV * GQA + hkv * GQA)) * DHEAD;

    // ---- Q A-fragments, resident for whole kernel -------------------------
    // A 16x4 layout: lane<16 holds (K=4kk, 4kk+1); lane>=16 holds (4kk+2, 4kk+3)
    // rows 0..3 = the 4 GQA heads (pre-scaled); rows 4..15 forced to zero.
    v2f aq[32];
    {
        const float rs = (lrow < GQA) ? scale : 0.0f;
        const float* qp = qb + (size_t)(lrow & 3) * DHEAD + 2 * half;
        #pragma unroll
        for (int kk = 0; kk < 32; ++kk) {
            v2f t = *(const v2f*)(qp + 4 * kk);
            aq[kk] = t * rs;
        }
    }

    const v8f vzero = {0.f, 0.f, 0.f, 0.f, 0.f, 0.f, 0.f, 0.f};
    v8f acc[8];
    #pragma unroll
    for (int t = 0; t < 8; ++t) acc[t] = vzero;

    float m0 = -1e30f, m1 = -1e30f, m2 = -1e30f, m3 = -1e30f;
    float l0 = 0.f, l1 = 0.f, l2 = 0.f, l3 = 0.f;

    const int key0base = w * CHUNK;
    const int hh = lane & 3;

    for (int tile = 0; tile < NTILES; ++tile) {
        const int key0 = key0base + tile * TILE;
        // per-lane K fragment base: row = key0+lrow, d offset 2*half
        const float* Kt = Kb + (size_t)(key0 + lrow) * DHEAD + 2 * half;
        // per-lane V fragment base: rows key0 + {0,1}(+2*half), col = lrow
        const float* Vt = Vb + (size_t)(key0 + 2 * half) * DHEAD + lrow;

        // prefetch next tile (global_prefetch_b8)
        __builtin_prefetch(Kt + TILE * DHEAD, 0, 3);
        __builtin_prefetch(Vt + TILE * DHEAD, 0, 3);

        // ---- scores = Q * K^T : 32 K-steps, two accumulator chains --------
        v8f s0 = vzero, s1 = vzero;
        #pragma unroll
        for (int kk = 0; kk < 32; kk += 2) {
            v2f kb0 = *(const v2f*)(Kt + 4 * kk);
            v2f kb1 = *(const v2f*)(Kt + 4 * kk + 4);
            s0 = __builtin_amdgcn_wmma_f32_16x16x4_f32(
                     false, aq[kk],     false, kb0, (short)0, s0, false, false);
            s1 = __builtin_amdgcn_wmma_f32_16x16x4_f32(
                     false, aq[kk + 1], false, kb1, (short)0, s1, false, false);
        }
        v8f sv = s0 + s1;

        // ---- online softmax over this 16-key tile -------------------------
        // lane n (n<16) holds score of key key0+n for head j in sv[j]
        float sj0 = sv[0], sj1 = sv[1], sj2 = sv[2], sj3 = sv[3];

        float t0 = sj0, t1 = sj1, t2 = sj2, t3 = sj3;
        #pragma unroll
        for (int mask = 1; mask <= 8; mask <<= 1) {
            t0 = fmaxf(t0, __shfl_xor(t0, mask, 32));
            t1 = fmaxf(t1, __shfl_xor(t1, mask, 32));
            t2 = fmaxf(t2, __shfl_xor(t2, mask, 32));
            t3 = fmaxf(t3, __shfl_xor(t3, mask, 32));
        }
        const float nm0 = fmaxf(m0, t0), nm1 = fmaxf(m1, t1);
        const float nm2 = fmaxf(m2, t2), nm3 = fmaxf(m3, t3);
        const float al0 = __expf(m0 - nm0), al1 = __expf(m1 - nm1);
        const float al2 = __expf(m2 - nm2), al3 = __expf(m3 - nm3);
        const float p0 = __expf(sj0 - nm0), p1 = __expf(sj1 - nm1);
        const float p2 = __expf(sj2 - nm2), p3 = __expf(sj3 - nm3);

        float r0 = p0, r1 = p1, r2 = p2, r3 = p3;
        #pragma unroll
        for (int mask = 1; mask <= 8; mask <<= 1) {
            r0 += __shfl_xor(r0, mask, 32);
            r1 += __shfl_xor(r1, mask, 32);
            r2 += __shfl_xor(r2, mask, 32);
            r3 += __shfl_xor(r3, mask, 32);
        }
        l0 = l0 * al0 + r0;  m0 = nm0;
        l1 = l1 * al1 + r1;  m1 = nm1;
        l2 = l2 * al2 + r2;  m2 = nm2;
        l3 = l3 * al3 + r3;  m3 = nm3;

        // rescale running output (rows 0..3 only; rows 4..15 are don't-care)
        #pragma unroll
        for (int t = 0; t < 8; ++t) {
            acc[t][0] *= al0; acc[t][1] *= al1;
            acc[t][2] *= al2; acc[t][3] *= al3;
        }

        // ---- transpose P into A layout via wave-private LDS ---------------
        if (half == 0) {
            P_lds[w][0][lrow] = p0;
            P_lds[w][1][lrow] = p1;
            P_lds[w][2][lrow] = p2;
            P_lds[w][3][lrow] = p3;
        }
        // same-wave LDS ops are in-order: load-after-store is safe

        // ---- O += P * V ----------------------------------------------------
        #pragma unroll
        for (int c = 0; c < 4; ++c) {
            // A frag: row = head (dup mod 4 for rows>=4, harmless), K = keys 4c..4c+3
            v2f pa = *(const v2f*)&P_lds[w][hh][4 * c + 2 * half];
            const float* Vc = Vt + (size_t)(4 * c) * DHEAD;
            #pragma unroll
            for (int t = 0; t < 8; ++t) {
                v2f vb;
                vb.x = Vc[16 * t];          // V[key0+4c+2*half  ][16t+lrow]
                vb.y = Vc[16 * t + DHEAD];  // V[key0+4c+2*half+1][16t+lrow]
                acc[t] = __builtin_amdgcn_wmma_f32_16x16x4_f32(
                             false, pa, false, vb, (short)0, acc[t], false, false);
            }
        }
    }

    // ---- per-wave partials to LDS -----------------------------------------
    if (half == 0) {
        #pragma unroll
        for (int t = 0; t < 8; ++t) {
            const int d = 16 * t + lrow;
            O_lds[w][0][d] = acc[t][0];
            O_lds[w][1][d] = acc[t][1];
            O_lds[w][2][d] = acc[t][2];
            O_lds[w][3][d] = acc[t][3];
        }
    }
    if (lane == 0) {
        m_lds[w][0] = m0; m_lds[w][1] = m1; m_lds[w][2] = m2; m_lds[w][3] = m3;
        l_lds[w][0] = l0; l_lds[w][1] = l1; l_lds[w][2] = l2; l_lds[w][3] = l3;
    }
    __syncthreads();

    // ---- combine 8 wave-partials (flash-decoding merge) -------------------
    for (int e = tid; e < GQA * DHEAD; e += 256) {
        const int j = e >> 7;
        const int d = e & (DHEAD - 1);
        float M = -1e30f;
        #pragma unroll
        for (int w2 = 0; w2 < NWAVES; ++w2) M = fmaxf(M, m_lds[w2][j]);
        float den = 0.f, num = 0.f;
        #pragma unroll
        for (int w2 = 0; w2 < NWAVES; ++w2) {
            const float f = __expf(m_lds[w2][j] - M);
            den += f * l_lds[w2][j];
            num += f * O_lds[w2][j][d];
        }
        out[((size_t)(b * (HKV * GQA) + hkv * GQA + j)) * DHEAD + d] = num / den;
    }
}

extern "C" void kernel_launch(void* const* d_in, const int* in_sizes, int n_in,
                              void* d_out, int out_size, void* d_ws, size_t ws_size,
                              hipStream_t stream) {
    (void)in_sizes; (void)n_in; (void)d_ws; (void)ws_size; (void)out_size;
    const float* q = (const float*)d_in[0];
    const float* K = (const float*)d_in[1];
    const float* V = (const float*)d_in[2];
    float* out = (float*)d_out;
    dim3 grid(32 * HKV);   // one block per (b, hkv) pair
    dim3 block(256);       // 8 waves; each wave owns 512 keys and 4 heads
    hipLaunchKernelGGL(gqa_decode_fused, grid, block, 0, stream, q, K, V, out);
}